// ChunkedMultiHeadCardPassingLayer_24068996727158
// MI455X (gfx1250) — compile-verified
//
#include <hip/hip_runtime.h>
#include <hip/hip_bf16.h>
#include <math.h>

typedef __attribute__((ext_vector_type(2))) float v2f;
typedef __attribute__((ext_vector_type(8))) float v8f;
typedef __attribute__((ext_vector_type(4))) unsigned u32x4;
typedef __attribute__((ext_vector_type(4))) int i32x4;
typedef __attribute__((ext_vector_type(8))) int i32x8;

#define B_   4
#define T_   4096
#define C_   1024
#define H_   16
#define CS_  128
#define D_   64
#define NC_  32
#define BT_  (B_ * T_)
#define EPS_ 1e-5f

// ---- CDNA5 fp32 WMMA: D = A(16x4) * B(4x16) + C(16x16) ----
__device__ __forceinline__ v8f wmma4(v2f a, v2f b, v8f c) {
  return __builtin_amdgcn_wmma_f32_16x16x4_f32(false, a, false, b, (short)0, c,
                                               false, false);
}

// A fragment (16x4) at p = &A[row0][k0], row-major, leading dim ld.
__device__ __forceinline__ v2f ldA(const float* p, int ld, int lane) {
  int m = lane & 15, k = (lane >> 4) << 1;
  v2f a;
  a.x = p[m * ld + k];
  a.y = p[m * ld + k + 1];
  return a;
}

// B fragment (4x16) at p = &B[k0][n0], row-major, leading dim ld.
__device__ __forceinline__ v2f ldB(const float* p, int ld, int lane) {
  int n = lane & 15, k = (lane >> 4) << 1;
  v2f b;
  b.x = p[k * ld + n];
  b.y = p[(k + 1) * ld + n];
  return b;
}

// ---- Tensor Data Mover: DMA a 2-D tile (tile_h rows x tile_w f32) from
// global (row stride stride_elems) into contiguous LDS.  D# per ISA 8.3-8.6.
// Issue from ONE wave (wave-uniform branch); track with TENSORcnt.
__device__ __forceinline__ void tdm_load_2d(const void* gsrc, void* ldst,
                                            int tile_w, int tile_h,
                                            int stride_elems) {
  unsigned long long ga = (unsigned long long)(uintptr_t)gsrc;
  unsigned lds_off = (unsigned)(uintptr_t)ldst;  // low 32b of LDS generic addr
  u32x4 g0;
  g0[0] = 1u;                                     // count=1 (valid), user mode
  g0[1] = lds_off;                                // lds_addr [63:32]
  g0[2] = (unsigned)ga;                           // global_addr [95:64]
  g0[3] = (unsigned)((ga >> 32) & 0x01FFFFFFu)    // global_addr [120:96]
          | (2u << 30);                           // type=2 ("image")
  unsigned td0 = (unsigned)stride_elems;          // tensor width (no OOB)
  unsigned td1 = 1u << 20;                        // plenty of rows
  i32x8 g1;
  g1[0] = (int)(2u << 16);                        // wg_mask=0, data_size=4B
  g1[1] = (int)((td0 & 0xFFFFu) << 16);           // tensor_dim0 lo16 @ [63:48]
  g1[2] = (int)(((td0 >> 16) & 0xFFFFu) |         // tensor_dim0 hi16
                ((td1 & 0xFFFFu) << 16));         // tensor_dim1 lo16
  g1[3] = (int)(((td1 >> 16) & 0xFFFFu) |         // tensor_dim1 hi16
                (((unsigned)tile_w & 0xFFFFu) << 16));  // tile_dim0
  g1[4] = (int)((unsigned)tile_h & 0xFFFFu);      // tile_dim1 (tile_dim2=0)
  g1[5] = (int)(unsigned)stride_elems;            // tensor_dim0_stride lo32
  g1[6] = 0;                                      // stride hi / dim1_stride
  g1[7] = 0;
  i32x4 gz = {0, 0, 0, 0};                        // 2-D: groups 2/3 unused
#if defined(__clang_major__) && (__clang_major__ >= 23)
  i32x8 gz8 = {0, 0, 0, 0, 0, 0, 0, 0};
  __builtin_amdgcn_tensor_load_to_lds(g0, g1, gz, gz, gz8, 0);
#else
  __builtin_amdgcn_tensor_load_to_lds(g0, g1, gz, gz, 0);
#endif
}

// =====================================================================
// K1: gm = sigmoid(x@gate_W + gate_b) * (x@mark_W + mark_b)
// 128x128 block tile; TDM double-buffered A/W tiles (96 KB LDS);
// wave0 issues DMA for tile k+1 while all 8 waves WMMA on tile k.
// =====================================================================
__global__ __launch_bounds__(256) void k_markgate(
    const float* __restrict__ x, const float* __restrict__ mW,
    const float* __restrict__ mb, const float* __restrict__ gW,
    const float* __restrict__ gb, float* __restrict__ gm) {
  __shared__ __align__(16) float As[2][128 * 32];
  __shared__ __align__(16) float Wm[2][32 * 128];
  __shared__ __align__(16) float Wg[2][32 * 128];
  const int tid = threadIdx.x, lane = tid & 31, w = tid >> 5;
  const int n0 = blockIdx.x * 128, m0 = blockIdx.y * 128;
  const int hl = lane >> 4, nl = lane & 15;

  v8f am[8], ag[8];
#pragma unroll
  for (int i = 0; i < 8; ++i) {
#pragma unroll
    for (int j = 0; j < 8; ++j) { am[i][j] = 0.f; ag[i][j] = 0.f; }
  }

  if (w == 0) {  // prologue DMA for k0 = 0
    tdm_load_2d(&x[(size_t)m0 * C_], &As[0][0], 32, 128, C_);
    tdm_load_2d(&mW[n0], &Wm[0][0], 128, 32, C_);
    tdm_load_2d(&gW[n0], &Wg[0][0], 128, 32, C_);
  }

  int cur = 0;
  for (int k0 = 0; k0 < C_; k0 += 32) {
    int nxt = cur ^ 1;
    if (w == 0) {
      if (k0 + 32 < C_) {
        tdm_load_2d(&x[(size_t)m0 * C_ + k0 + 32], &As[nxt][0], 32, 128, C_);
        tdm_load_2d(&mW[(size_t)(k0 + 32) * C_ + n0], &Wm[nxt][0], 128, 32, C_);
        tdm_load_2d(&gW[(size_t)(k0 + 32) * C_ + n0], &Wg[nxt][0], 128, 32, C_);
        __builtin_amdgcn_s_wait_tensorcnt(3);  // in-order: cur's 3 complete
      } else {
        __builtin_amdgcn_s_wait_tensorcnt(0);
      }
    }
    __syncthreads();  // cur tile ready for everyone
    const float* Asb = &As[cur][0];
    const float* Wmb = &Wm[cur][0];
    const float* Wgb = &Wg[cur][0];
#pragma unroll
    for (int kk = 0; kk < 32; kk += 4) {
      v2f a = ldA(&Asb[(w * 16) * 32 + kk], 32, lane);
#pragma unroll
      for (int nt = 0; nt < 8; ++nt) {
        v2f bm = ldB(&Wmb[kk * 128 + nt * 16], 128, lane);
        am[nt] = wmma4(a, bm, am[nt]);
        v2f bg = ldB(&Wgb[kk * 128 + nt * 16], 128, lane);
        ag[nt] = wmma4(a, bg, ag[nt]);
      }
    }
    __syncthreads();  // done reading cur before TDM overwrites it
    cur = nxt;
  }

#pragma unroll
  for (int nt = 0; nt < 8; ++nt) {
    int col = n0 + nt * 16 + nl;
    float mbv = mb[col], gbv = gb[col];
#pragma unroll
    for (int r = 0; r < 8; ++r) {
      int row = m0 + w * 16 + r + hl * 8;
      float pm = am[nt][r] + mbv;
      float gt = ag[nt][r] + gbv;
      float sg = 1.f / (1.f + __expf(-gt));
      gm[(size_t)row * C_ + col] = sg * pm;
    }
  }
}

// =====================================================================
// K2: in-place cumsum of gm over the 128 positions of each chunk
// =====================================================================
__global__ __launch_bounds__(64) void k_cumsum(float* __restrict__ gm,
                                               float* __restrict__ csum) {
  const int d = threadIdx.x;
  const int chunk = blockIdx.x, h = blockIdx.y, b = blockIdx.z;
  size_t base = ((size_t)b * T_ + (size_t)chunk * CS_) * C_ + h * D_ + d;
  float acc = 0.f;
  for (int s = 0; s < CS_; ++s) {
    size_t idx = base + (size_t)s * C_;
    acc += gm[idx];
    gm[idx] = acc;
  }
  csum[(((size_t)b * H_ + h) * NC_ + chunk) * D_ + d] = acc;
}

// =====================================================================
// K3: carries = exclusive prefix over chunks; nrm_carry = LN_D(carries)
// =====================================================================
__global__ __launch_bounds__(64) void k_carry_ln(
    const float* __restrict__ csum, const float* __restrict__ cg,
    const float* __restrict__ cb, float* __restrict__ ncar) {
  __shared__ float s1[64], s2[64];
  const int d = threadIdx.x;
  const int chunk = blockIdx.x, h = blockIdx.y, b = blockIdx.z;
  const size_t bh = (size_t)b * H_ + h;
  float carry = 0.f;
  for (int j = 0; j < chunk; ++j) carry += csum[(bh * NC_ + j) * D_ + d];
  s1[d] = carry;
  s2[d] = carry * carry;
  __syncthreads();
  for (int off = 32; off > 0; off >>= 1) {
    if (d < off) { s1[d] += s1[d + off]; s2[d] += s2[d + off]; }
    __syncthreads();
  }
  float m = s1[0] * (1.f / 64.f);
  float v = s2[0] * (1.f / 64.f) - m * m;
  float rs = rsqrtf(v + EPS_);
  ncar[(bh * NC_ + chunk) * D_ + d] = (carry - m) * rs * cg[d] + cb[d];
}

// =====================================================================
// K4: per (b,h,chunk): comb=[x_h | LN(cards)] (64 KB LDS), W1/W2 TDM'd
// into LDS (96 KB) overlapped with cards construction; two WMMA GEMMs.
// =====================================================================
__global__ __launch_bounds__(256) void k_headmlp(
    const float* __restrict__ x, const float* __restrict__ lcm,
    const float* __restrict__ ncar, const float* __restrict__ cdg,
    const float* __restrict__ cdb, const float* __restrict__ W1,
    const float* __restrict__ b1, const float* __restrict__ W2,
    const float* __restrict__ b2, float* __restrict__ ho) {
  __shared__ __align__(16) float comb[128 * 128];  // 64 KB
  __shared__ __align__(16) float W1s[128 * 128];   // 64 KB
  __shared__ __align__(16) float W2s[128 * 64];    // 32 KB
  const int tid = threadIdx.x, lane = tid & 31, w = tid >> 5;
  const int chunk = blockIdx.x, h = blockIdx.y, b = blockIdx.z;
  const int hl = lane >> 4, nl = lane & 15;
  const size_t rowbase = ((size_t)b * T_ + (size_t)chunk * CS_) * C_ + h * D_;
  const size_t nbase = (((size_t)b * H_ + h) * NC_ + chunk) * D_;

  if (w == 0) {  // DMA the head-MLP weights; overlaps with phase A below
    tdm_load_2d(W1, &W1s[0], 128, 128, 128);
    tdm_load_2d(W2, &W2s[0], 64, 128, 64);
  }

  if (tid < 128) {
    const int s = tid;
#pragma unroll
    for (int c4 = 0; c4 < 16; ++c4)
      *(float4*)&comb[s * 128 + c4 * 4] =
          *(const float4*)&x[rowbase + (size_t)s * C_ + c4 * 4];
    float cl[64];
    float sum = 0.f, sq = 0.f;
#pragma unroll
    for (int d = 0; d < 64; ++d) {
      float nv = ncar[nbase + d];
      float val = (s == 0) ? nv : (lcm[rowbase + (size_t)(s - 1) * C_ + d] + nv);
      cl[d] = val;
      sum += val;
      sq += val * val;
    }
    float m = sum * (1.f / 64.f);
    float v = sq * (1.f / 64.f) - m * m;
    float rs = rsqrtf(v + EPS_);
#pragma unroll
    for (int d = 0; d < 64; ++d)
      comb[s * 128 + 64 + d] = (cl[d] - m) * rs * cdg[d] + cdb[d];
  }
  if (w == 0) __builtin_amdgcn_s_wait_tensorcnt(0);
  __syncthreads();

  // GEMM1: rows [w*16, w*16+16) x 128 cols, K=128
  v8f acc[8];
#pragma unroll
  for (int i = 0; i < 8; ++i) {
#pragma unroll
    for (int j = 0; j < 8; ++j) acc[i][j] = 0.f;
  }
  for (int k = 0; k < 128; k += 4) {
    v2f a = ldA(&comb[(w * 16) * 128 + k], 128, lane);
#pragma unroll
    for (int nt = 0; nt < 8; ++nt) {
      v2f bf = ldB(&W1s[k * 128 + nt * 16], 128, lane);
      acc[nt] = wmma4(a, bf, acc[nt]);
    }
  }
  __syncthreads();  // everyone done reading comb
#pragma unroll
  for (int nt = 0; nt < 8; ++nt) {
    int col = nt * 16 + nl;
    float bias = b1[col];
#pragma unroll
    for (int r = 0; r < 8; ++r) {
      float vv = acc[nt][r] + bias;
      vv = 0.5f * vv * (1.f + erff(vv * 0.70710678118f));  // exact gelu
      comb[(w * 16 + r + hl * 8) * 128 + col] = vv;
    }
  }
  __syncthreads();

  // GEMM2: rows strip x 64 cols, K=128
  v8f acc2[4];
#pragma unroll
  for (int i = 0; i < 4; ++i) {
#pragma unroll
    for (int j = 0; j < 8; ++j) acc2[i][j] = 0.f;
  }
  for (int k = 0; k < 128; k += 4) {
    v2f a = ldA(&comb[(w * 16) * 128 + k], 128, lane);
#pragma unroll
    for (int nt = 0; nt < 4; ++nt) {
      v2f bf = ldB(&W2s[k * 64 + nt * 16], 64, lane);
      acc2[nt] = wmma4(a, bf, acc2[nt]);
    }
  }
#pragma unroll
  for (int nt = 0; nt < 4; ++nt) {
    int col = nt * 16 + nl;
    float bias = b2[col];
#pragma unroll
    for (int r = 0; r < 8; ++r) {
      int s = w * 16 + r + hl * 8;
      ho[rowbase + (size_t)s * C_ + col] = acc2[nt][r] + bias;
    }
  }
}

// =====================================================================
// K5: out = x + LN_C(ho @ proj_W + proj_b)
// Block = 16 rows x full 1024 cols so the row LN fuses in-kernel.
// =====================================================================
__global__ __launch_bounds__(256) void k_proj_ln(
    const float* __restrict__ ho, const float* __restrict__ pW,
    const float* __restrict__ pb, const float* __restrict__ lng,
    const float* __restrict__ lnb, const float* __restrict__ x,
    float* __restrict__ out) {
  __shared__ float s_sum[16], s_sq[16], s_mean[16], s_rstd[16];
  const int tid = threadIdx.x, lane = tid & 31, w = tid >> 5;
  const int m0 = blockIdx.x * 16;
  const int hl = lane >> 4, nl = lane & 15;
  if (tid < 16) { s_sum[tid] = 0.f; s_sq[tid] = 0.f; }

  v8f acc[8];
#pragma unroll
  for (int i = 0; i < 8; ++i) {
#pragma unroll
    for (int j = 0; j < 8; ++j) acc[i][j] = 0.f;
  }
  for (int k = 0; k < C_; k += 4) {
    v2f a = ldA(&ho[(size_t)m0 * C_ + k], C_, lane);
#pragma unroll
    for (int nt = 0; nt < 8; ++nt) {
      v2f bf = ldB(&pW[(size_t)k * C_ + w * 128 + nt * 16], C_, lane);
      acc[nt] = wmma4(a, bf, acc[nt]);
    }
  }

  float p[8], q[8];
#pragma unroll
  for (int r = 0; r < 8; ++r) { p[r] = 0.f; q[r] = 0.f; }
#pragma unroll
  for (int nt = 0; nt < 8; ++nt) {
    int col = w * 128 + nt * 16 + nl;
    float bias = pb[col];
#pragma unroll
    for (int r = 0; r < 8; ++r) {
      float vv = acc[nt][r] + bias;
      acc[nt][r] = vv;
      p[r] += vv;
      q[r] += vv * vv;
    }
  }
  __syncthreads();  // s_sum/s_sq init visible
#pragma unroll
  for (int r = 0; r < 8; ++r) {
    atomicAdd(&s_sum[r + hl * 8], p[r]);  // ds_add_f32
    atomicAdd(&s_sq[r + hl * 8], q[r]);
  }
  __syncthreads();
  if (tid < 16) {
    float m = s_sum[tid] * (1.f / (float)C_);
    float v = s_sq[tid] * (1.f / (float)C_) - m * m;
    s_mean[tid] = m;
    s_rstd[tid] = rsqrtf(v + EPS_);
  }
  __syncthreads();
#pragma unroll
  for (int nt = 0; nt < 8; ++nt) {
    int col = w * 128 + nt * 16 + nl;
    float g = lng[col], bb = lnb[col];
#pragma unroll
    for (int r = 0; r < 8; ++r) {
      int row = r + hl * 8;
      size_t gidx = (size_t)(m0 + row) * C_ + col;
      float o = (acc[nt][r] - s_mean[row]) * s_rstd[row] * g + bb;
      out[gidx] = x[gidx] + o;
    }
  }
}

// =====================================================================
extern "C" void kernel_launch(void* const* d_in, const int* in_sizes, int n_in,
                              void* d_out, int out_size, void* d_ws,
                              size_t ws_size, hipStream_t stream) {
  const float* x       = (const float*)d_in[0];
  const float* mark_W  = (const float*)d_in[1];
  const float* mark_b  = (const float*)d_in[2];
  const float* gate_W  = (const float*)d_in[3];
  const float* gate_b  = (const float*)d_in[4];
  const float* carry_g = (const float*)d_in[5];
  const float* carry_b = (const float*)d_in[6];
  const float* card_g  = (const float*)d_in[7];
  const float* card_b  = (const float*)d_in[8];
  const float* ho1_W   = (const float*)d_in[9];
  const float* ho1_b   = (const float*)d_in[10];
  const float* ho2_W   = (const float*)d_in[11];
  const float* ho2_b   = (const float*)d_in[12];
  const float* proj_W  = (const float*)d_in[13];
  const float* proj_b  = (const float*)d_in[14];
  const float* ln_g    = (const float*)d_in[15];
  const float* ln_b    = (const float*)d_in[16];
  float* out = (float*)d_out;

  // workspace layout (floats): gm/lcm | chunk_sums | nrm_carry | ho
  float* buf0 = (float*)d_ws;                       // BT*C (gm -> lcm in place)
  float* buf1 = buf0 + (size_t)BT_ * C_;            // B*H*NC*D
  float* buf2 = buf1 + (size_t)B_ * H_ * NC_ * D_;  // B*H*NC*D
  float* buf3 = buf2 + (size_t)B_ * H_ * NC_ * D_;  // BT*C

  k_markgate<<<dim3(C_ / 128, BT_ / 128), 256, 0, stream>>>(
      x, mark_W, mark_b, gate_W, gate_b, buf0);
  k_cumsum<<<dim3(NC_, H_, B_), 64, 0, stream>>>(buf0, buf1);
  k_carry_ln<<<dim3(NC_, H_, B_), 64, 0, stream>>>(buf1, carry_g, carry_b,
                                                   buf2);
  k_headmlp<<<dim3(NC_, H_, B_), 256, 0, stream>>>(
      x, buf0, buf2, card_g, card_b, ho1_W, ho1_b, ho2_W, ho2_b, buf3);
  k_proj_ln<<<dim3(BT_ / 16), 256, 0, stream>>>(buf3, proj_W, proj_b, ln_g,
                                                ln_b, x, out);
}